// AttnDecoderRNN_53824530154057
// MI455X (gfx1250) — compile-verified
//
#include <hip/hip_runtime.h>
#include <hip/hip_bf16.h>

// Problem constants (match reference)
#define BB 128
#define SS 512
#define HH 1024

typedef __attribute__((ext_vector_type(2))) float v2f;
typedef __attribute__((ext_vector_type(8))) float v8f;

// ---------------------------------------------------------------------------
// fp32 WMMA GEMM: C[M,N] = act( A[M,K] * W^T + bias )
//   WT=true : W stored [N,K] row-major ("NT")
//   WT=false: W stored [K,N] row-major ("NN")
// One wave computes a 16x64 strip of C (4 tiles of 16x16) with
// V_WMMA_F32_16X16X4_F32; the A fragment is reused across the 4 N-tiles so
// A-side VMEM traffic is amortized 4x and 4 independent WMMAs are in flight.
// EXEC is all 1s at every WMMA (blockDim=32, no divergence, exact grids).
// act: 0 = none, 1 = tanh, 2 = sigmoid
// ---------------------------------------------------------------------------
template <bool WT>
__global__ __launch_bounds__(32)
void gemm_wmma_f32(const float* __restrict__ A, const float* __restrict__ W,
                   const float* __restrict__ bias, float* __restrict__ C,
                   int M, int N, int K, int act) {
  const int lane = threadIdx.x & 31;
  const int half = lane >> 4;   // 0: lanes 0-15, 1: lanes 16-31
  const int l    = lane & 15;
  const int n0   = blockIdx.x * 64;   // 4 x 16 columns per wave
  const int m0   = blockIdx.y * 16;

  // fp32 A 16x4 fragment: VGPR0 = K(0|2), VGPR1 = K(1|3) per lane-half.
  // fp32 B 4x16 fragment mirrors it (rows striped across lanes).
  const float* __restrict__ arow = A + (size_t)(m0 + l) * K;

  v8f acc[4] = {{}, {}, {}, {}};
  for (int k = 0; k < K; k += 4) {
    const int kk = k + 2 * half;
    v2f a;
    a.x = arow[kk];
    a.y = arow[kk + 1];

    v2f b[4];
    if (WT) {
      const float* __restrict__ w = W + (size_t)(n0 + l) * K + kk;
#pragma unroll
      for (int t = 0; t < 4; ++t) {
        b[t].x = w[(size_t)(16 * t) * K];
        b[t].y = w[(size_t)(16 * t) * K + 1];
      }
    } else {
      const float* __restrict__ w = W + (size_t)kk * N + n0 + l;
#pragma unroll
      for (int t = 0; t < 4; ++t) {
        b[t].x = w[16 * t];
        b[t].y = w[16 * t + N];
      }
    }

#pragma unroll
    for (int t = 0; t < 4; ++t) {
      acc[t] = __builtin_amdgcn_wmma_f32_16x16x4_f32(
          /*neg_a=*/false, a, /*neg_b=*/false, b[t],
          /*c_mod=*/(short)0, acc[t], /*reuse_a=*/false, /*reuse_b=*/false);
    }
  }

#pragma unroll
  for (int t = 0; t < 4; ++t) {
    const int n = n0 + 16 * t + l;
    const float bs = bias ? bias[n] : 0.0f;
#pragma unroll
    for (int r = 0; r < 8; ++r) {
      float v = acc[t][r] + bs;
      if (act == 1)      v = tanhf(v);
      else if (act == 2) v = 1.0f / (1.0f + __expf(-v));
      C[(size_t)(m0 + r + 8 * half) * N + n] = v;
    }
  }
}

// ---------------------------------------------------------------------------
// GRU gate fusion: r,z,n gates + new hidden state. Writes rnn_output to ws and
// the `hidden` output slot of d_out (identical [B,H] layout).
// ---------------------------------------------------------------------------
__global__ void gru_gates_kernel(const float* __restrict__ gi,
                                 const float* __restrict__ gh,
                                 const float* __restrict__ h,
                                 float* __restrict__ rnn,
                                 float* __restrict__ hidden_out) {
  const int idx = blockIdx.x * blockDim.x + threadIdx.x;  // [0, B*H)
  const int b = idx >> 10;       // H = 1024
  const int j = idx & (HH - 1);
  const float* gib = gi + (size_t)b * 3 * HH;
  const float* ghb = gh + (size_t)b * 3 * HH;

  const float ir = gib[j],           hr = ghb[j];
  const float iz = gib[HH + j],      hz = ghb[HH + j];
  const float in_ = gib[2 * HH + j], hn = ghb[2 * HH + j];

  const float r = 1.0f / (1.0f + __expf(-(ir + hr)));
  const float z = 1.0f / (1.0f + __expf(-(iz + hz)));
  const float n = tanhf(in_ + r * hn);
  const float o = (1.0f - z) * n + z * h[idx];

  rnn[idx] = o;
  hidden_out[idx] = o;
}

// ---------------------------------------------------------------------------
// energies[b,s] = dot(q[b,:], enc[s,b,:]) — one wave32 per (b,s) pair.
// float4 (b128) loads: each wave streams its 4 KB enc row in 8 iterations.
// (The rnn.b_attn term is constant over s and cancels in the softmax.)
// ---------------------------------------------------------------------------
__global__ void energies_kernel(const float* __restrict__ q,
                                const float* __restrict__ enc,
                                float* __restrict__ energies) {
  const int wid  = (int)((blockIdx.x * blockDim.x + threadIdx.x) >> 5);
  const int lane = threadIdx.x & 31;
  const int b = wid >> 9;        // S = 512
  const int s = wid & (SS - 1);

  const float4* __restrict__ e4 =
      (const float4*)(enc + ((size_t)s * BB + b) * HH);
  const float4* __restrict__ q4 = (const float4*)(q + (size_t)b * HH);

  float acc = 0.0f;
#pragma unroll
  for (int d = lane; d < HH / 4; d += 32) {
    const float4 qa = q4[d];
    const float4 ea = e4[d];
    acc += qa.x * ea.x + qa.y * ea.y + qa.z * ea.z + qa.w * ea.w;
  }
#pragma unroll
  for (int off = 16; off > 0; off >>= 1) acc += __shfl_xor(acc, off, 32);
  if (lane == 0) energies[(size_t)b * SS + s] = acc;
}

// ---------------------------------------------------------------------------
// Row softmax over S=512, one 256-thread block per batch row. Writes both the
// attn_weights output slice of d_out and a workspace copy for the context pass.
// ---------------------------------------------------------------------------
__global__ void softmax_kernel(const float* __restrict__ energies,
                               float* __restrict__ attn_ws,
                               float* __restrict__ attn_out) {
  __shared__ float red[256];
  const int b = blockIdx.x;
  const int t = threadIdx.x;
  const float* __restrict__ e = energies + (size_t)b * SS;

  float m = -1e30f;
  for (int s = t; s < SS; s += 256) m = fmaxf(m, e[s]);
  red[t] = m; __syncthreads();
  for (int w = 128; w > 0; w >>= 1) {
    if (t < w) red[t] = fmaxf(red[t], red[t + w]);
    __syncthreads();
  }
  m = red[0]; __syncthreads();

  float sum = 0.0f;
  for (int s = t; s < SS; s += 256) sum += __expf(e[s] - m);
  red[t] = sum; __syncthreads();
  for (int w = 128; w > 0; w >>= 1) {
    if (t < w) red[t] += red[t + w];
    __syncthreads();
  }
  const float inv = 1.0f / red[0];

  for (int s = t; s < SS; s += 256) {
    const float p = __expf(e[s] - m) * inv;
    attn_ws[(size_t)b * SS + s] = p;
    attn_out[(size_t)b * SS + s] = p;
  }
}

// ---------------------------------------------------------------------------
// context[b,:] = sum_s attn[b,s] * enc[s,b,:]. One 256-thread block per b;
// each thread owns 4 consecutive columns (float4), so per s the block loads a
// full contiguous 4 KB row with b128 accesses. attn row staged in LDS.
// Second (and last) streaming pass over enc; prefetch hints the stream.
// ---------------------------------------------------------------------------
__global__ void context_kernel(const float* __restrict__ attn,
                               const float* __restrict__ enc,
                               float* __restrict__ ctx) {
  __shared__ float aw[SS];
  const int b = blockIdx.x;
  const int t = threadIdx.x;   // 256 threads * float4 = 1024 columns

  for (int s = t; s < SS; s += 256) aw[s] = attn[(size_t)b * SS + s];
  __syncthreads();

  const float4* __restrict__ eb = (const float4*)(enc + (size_t)b * HH) + t;
  const size_t stride4 = (size_t)BB * HH / 4;   // float4 stride between s rows

  float4 acc = make_float4(0.f, 0.f, 0.f, 0.f);
  for (int s = 0; s < SS; ++s) {
    if (s + 4 < SS) __builtin_prefetch(&eb[(size_t)(s + 4) * stride4], 0, 0);
    const float4 ev = eb[(size_t)s * stride4];
    const float w = aw[s];
    acc.x += w * ev.x;
    acc.y += w * ev.y;
    acc.z += w * ev.z;
    acc.w += w * ev.w;
  }
  ((float4*)(ctx + (size_t)b * HH))[t] = acc;
}

// Pack [rnn | context] -> concat_input [B, 2H]
__global__ void concat_pack_kernel(const float* __restrict__ rnn,
                                   const float* __restrict__ ctx,
                                   float* __restrict__ cin) {
  const int idx = blockIdx.x * blockDim.x + threadIdx.x;  // [0, B*H)
  const int b = idx >> 10;
  const int j = idx & (HH - 1);
  cin[(size_t)b * 2 * HH + j] = rnn[idx];
  cin[(size_t)b * 2 * HH + HH + j] = ctx[idx];
}

// ---------------------------------------------------------------------------
extern "C" void kernel_launch(void* const* d_in, const int* in_sizes, int n_in,
                              void* d_out, int out_size, void* d_ws, size_t ws_size,
                              hipStream_t stream) {
  (void)in_sizes; (void)n_in; (void)out_size; (void)ws_size;

  const float* input_seq = (const float*)d_in[0];   // [B,H]
  const float* last_hid  = (const float*)d_in[1];   // [1,B,H]
  const float* enc       = (const float*)d_in[2];   // [S,B,H]
  const float* W_ih      = (const float*)d_in[3];   // [3H,H]
  const float* b_ih      = (const float*)d_in[4];
  const float* W_hh      = (const float*)d_in[5];   // [3H,H]
  const float* b_hh      = (const float*)d_in[6];
  const float* W_attn    = (const float*)d_in[7];   // [H,H]
  // d_in[8] = b_attn: constant over s per batch row -> cancels in softmax.
  const float* W_concat  = (const float*)d_in[9];   // [H,2H]
  const float* b_concat  = (const float*)d_in[10];
  const float* W_out     = (const float*)d_in[11];  // [H,H]
  const float* b_out     = (const float*)d_in[12];

  float* out    = (float*)d_out;            // output   [B,H]
  float* hidden = out + BB * HH;            // hidden   [1,B,H]
  float* attn_o = out + 2 * BB * HH;        // attn_wts [B,1,S]

  // Workspace layout (floats), with region reuse after the GRU step:
  float* ws   = (float*)d_ws;
  float* gi   = ws;                         // [B,3H]  393216
  float* gh   = ws + 393216;                // [B,3H]  393216
  float* rnn  = ws + 786432;                // [B,H]   131072
  //   gi region reused after gates:
  float* q    = ws;                         // [B,H]   131072
  float* en   = ws + 131072;                // [B,S]    65536
  float* at   = ws + 196608;                // [B,S]    65536
  float* ctx  = ws + 262144;                // [B,H]   131072
  //   gh region reused after gates:
  float* cin  = ws + 393216;                // [B,2H]  262144
  float* cout = ws + 655360;                // [B,H]   131072

  const dim3 wave(32);

  // --- GRU input/hidden GEMMs: [128 x 3072] = [128 x 1024] * W^T -----------
  gemm_wmma_f32<true><<<dim3(3 * HH / 64, BB / 16), wave, 0, stream>>>(
      input_seq, W_ih, b_ih, gi, BB, 3 * HH, HH, /*act=*/0);
  gemm_wmma_f32<true><<<dim3(3 * HH / 64, BB / 16), wave, 0, stream>>>(
      last_hid, W_hh, b_hh, gh, BB, 3 * HH, HH, /*act=*/0);

  // --- gates -> rnn_output, also writes `hidden` output --------------------
  gru_gates_kernel<<<BB * HH / 256, 256, 0, stream>>>(gi, gh, last_hid, rnn, hidden);

  // --- attention query: q = rnn @ W_attn (NN GEMM) --------------------------
  gemm_wmma_f32<false><<<dim3(HH / 64, BB / 16), wave, 0, stream>>>(
      rnn, W_attn, nullptr, q, BB, HH, HH, /*act=*/0);

  // --- energies: one wave per (b,s); first streaming pass over enc ----------
  energies_kernel<<<(BB * SS * 32) / 256, 256, 0, stream>>>(q, enc, en);

  // --- softmax over S per batch row -----------------------------------------
  softmax_kernel<<<BB, 256, 0, stream>>>(en, at, attn_o);

  // --- context: second streaming pass over enc ------------------------------
  context_kernel<<<BB, 256, 0, stream>>>(at, enc, ctx);

  // --- concat head: tanh(cin @ W_concat^T + b), sigmoid(... @ W_out^T + b) --
  concat_pack_kernel<<<BB * HH / 256, 256, 0, stream>>>(rnn, ctx, cin);
  gemm_wmma_f32<true><<<dim3(HH / 64, BB / 16), wave, 0, stream>>>(
      cin, W_concat, b_concat, cout, BB, HH, 2 * HH, /*act=*/1);
  gemm_wmma_f32<true><<<dim3(HH / 64, BB / 16), wave, 0, stream>>>(
      cout, W_out, b_out, out, BB, HH, HH, /*act=*/2);
}